// LightGCN_2894807958214
// MI455X (gfx1250) — compile-verified
//
#include <hip/hip_runtime.h>

// ---------------- problem constants (match reference) ----------------
#define NUM_USERS  100000
#define NUM_ITEMS  50000
#define NUM_NODES  (NUM_USERS + NUM_ITEMS)   // 150000
#define EMBED_DIM  128
#define NUM_LAYERS 3
#define NUM_EDGES  4800000

#define EDGES_PER_WAVE 512

typedef __attribute__((ext_vector_type(2))) float v2f;
typedef __attribute__((ext_vector_type(8))) float v8f;

static_assert((NUM_NODES * (long long)EMBED_DIM) % 256 == 0, "tileable");
static_assert(NUM_EDGES % EDGES_PER_WAVE == 0, "no edge tail");
static_assert(EDGES_PER_WAVE % 32 == 0, "chunk is whole waves of edges");

// ---------------------------------------------------------------------
// init: h0 = acc = concat(user_emb, item_emb)   (float4 vectorized)
// ---------------------------------------------------------------------
__global__ void lgcn_init_kernel(const float* __restrict__ user_emb,
                                 const float* __restrict__ item_emb,
                                 float* __restrict__ h0,
                                 float* __restrict__ acc) {
    size_t i = (size_t)blockIdx.x * blockDim.x + threadIdx.x;   // float4 index
    const size_t total4   = (size_t)NUM_NODES * EMBED_DIM / 4;  // 4.8M
    const size_t userEnd4 = (size_t)NUM_USERS * EMBED_DIM / 4;  // 3.2M
    if (i >= total4) return;
    float4 v = (i < userEnd4) ? ((const float4*)user_emb)[i]
                              : ((const float4*)item_emb)[i - userEnd4];
    ((float4*)h0)[i]  = v;
    ((float4*)acc)[i] = v;
}

// ---------------------------------------------------------------------
// zero a float buffer (float4 vectorized)
// ---------------------------------------------------------------------
__global__ void lgcn_zero_kernel(float* __restrict__ p, size_t n4) {
    size_t i = (size_t)blockIdx.x * blockDim.x + threadIdx.x;
    if (i >= n4) return;
    ((float4*)p)[i] = make_float4(0.f, 0.f, 0.f, 0.f);
}

// ---------------------------------------------------------------------
// SpMM: h_new[r] += val * h[c] over sorted-row COO edges.
// One wave per 512-edge chunk. Lane l owns dims [4l, 4l+4) (float4).
// Edge triples are loaded 32-at-a-time; the broadcast index j is a
// uniform loop counter, so v_readlane_b32 (not ds_bpermute) moves the
// triple into SGPRs -> row-run test is an SALU compare + scalar branch.
// Row runs (rows sorted) accumulate in registers; flush once per run
// with relaxed agent-scope f32 atomics (handles chunk-boundary splits).
// h (76.8 MB) fits in the 192 MB L2 -> gathers are L2 hits.
// ---------------------------------------------------------------------
__global__ void lgcn_spmm_kernel(const int*   __restrict__ rows,
                                 const int*   __restrict__ cols,
                                 const float* __restrict__ vals,
                                 const float* __restrict__ h,
                                 float*       __restrict__ hnew) {
    const int lane = threadIdx.x & 31;
    const long long wave = ((long long)blockIdx.x * blockDim.x + threadIdx.x) >> 5;
    const long long base = wave * EDGES_PER_WAVE;
    if (base >= NUM_EDGES) return;                 // whole-wave uniform exit
    const long long end = base + EDGES_PER_WAVE;   // E % 512 == 0 -> no tail

    const int d = lane * 4;                        // this lane's dim offset
    float4 acc = make_float4(0.f, 0.f, 0.f, 0.f);
    int cur = -1;

    for (long long e0 = base; e0 < end; e0 += 32) {
        const long long me = e0 + lane;            // always in range
        const int   r  = rows[me];
        const int   ci = cols[me];
        const float v  = vals[me];
#pragma unroll 4
        for (int j = 0; j < 32; ++j) {
            const int   rj = __builtin_amdgcn_readlane(r, j);          // SGPR
            const int   cj = __builtin_amdgcn_readlane(ci, j);         // SGPR
            const float vj = __uint_as_float(
                __builtin_amdgcn_readlane(__float_as_uint(v), j));     // SGPR
            if (rj != cur) {                       // scalar row-run boundary
                if (cur >= 0) {
                    float* dst = hnew + (size_t)cur * EMBED_DIM + d;
                    __hip_atomic_fetch_add(dst + 0, acc.x, __ATOMIC_RELAXED, __HIP_MEMORY_SCOPE_AGENT);
                    __hip_atomic_fetch_add(dst + 1, acc.y, __ATOMIC_RELAXED, __HIP_MEMORY_SCOPE_AGENT);
                    __hip_atomic_fetch_add(dst + 2, acc.z, __ATOMIC_RELAXED, __HIP_MEMORY_SCOPE_AGENT);
                    __hip_atomic_fetch_add(dst + 3, acc.w, __ATOMIC_RELAXED, __HIP_MEMORY_SCOPE_AGENT);
                }
                acc = make_float4(0.f, 0.f, 0.f, 0.f);
                cur = rj;
            }
            const float4 hv = *(const float4*)(h + (size_t)cj * EMBED_DIM + d);
            acc.x = fmaf(vj, hv.x, acc.x);
            acc.y = fmaf(vj, hv.y, acc.y);
            acc.z = fmaf(vj, hv.z, acc.z);
            acc.w = fmaf(vj, hv.w, acc.w);
        }
    }
    if (cur >= 0) {
        float* dst = hnew + (size_t)cur * EMBED_DIM + d;
        __hip_atomic_fetch_add(dst + 0, acc.x, __ATOMIC_RELAXED, __HIP_MEMORY_SCOPE_AGENT);
        __hip_atomic_fetch_add(dst + 1, acc.y, __ATOMIC_RELAXED, __HIP_MEMORY_SCOPE_AGENT);
        __hip_atomic_fetch_add(dst + 2, acc.z, __ATOMIC_RELAXED, __HIP_MEMORY_SCOPE_AGENT);
        __hip_atomic_fetch_add(dst + 3, acc.w, __ATOMIC_RELAXED, __HIP_MEMORY_SCOPE_AGENT);
    }
}

// ---------------------------------------------------------------------
// acc += h_new, and (optionally) zero the next layer's destination
// buffer in the same pass (saves a launch + a full write pass).
// ---------------------------------------------------------------------
__global__ void lgcn_add_zero_kernel(float* __restrict__ acc,
                                     const float* __restrict__ hnew,
                                     float* __restrict__ tozero,   // may be null
                                     size_t n4) {
    size_t i = (size_t)blockIdx.x * blockDim.x + threadIdx.x;
    if (i >= n4) return;
    float4 a = ((const float4*)acc)[i];
    float4 b = ((const float4*)hnew)[i];
    a.x += b.x; a.y += b.y; a.z += b.z; a.w += b.w;
    ((float4*)acc)[i] = a;
    if (tozero) ((float4*)tozero)[i] = make_float4(0.f, 0.f, 0.f, 0.f);
}

// ---------------------------------------------------------------------
// finalize: out *= 0.25, done on the matrix pipe (exact in f32).
// One wave per 256-float tile T (16x16 row-major).
//   0.25*T = sum_k (0.25*I16)[:,4k:4k+4] x T[4k:4k+4,:]
// 4 chained V_WMMA_F32_16X16X4_F32, D accumulates in the documented
// 16x16 f32 C/D layout (8 VGPRs). Grid sized exactly -> EXEC all-1s.
// A 16x4 f32 layout: lanes 0-15 M=0..15 hold K={0,1}; lanes 16-31 K={2,3}.
// B 4x16 f32 layout (symmetric): lanes 0-15 N, K={0,1}; lanes 16-31 K={2,3}.
// ---------------------------------------------------------------------
__global__ void lgcn_finalize_wmma_kernel(float* __restrict__ out) {
    const int lane = threadIdx.x & 31;
    const size_t tile = (size_t)blockIdx.x * (blockDim.x >> 5) + (threadIdx.x >> 5);
    float* p = out + tile * 256;

    const int m  = lane & 15;     // row (for A) / col n (for B and C/D)
    const int hi = lane >> 4;     // 0 -> K pair {0,1}; 1 -> K pair {2,3}

    v8f c = {};
#pragma unroll
    for (int k = 0; k < 4; ++k) {
        const int k0 = 4 * k + (hi ? 2 : 0);
        const int k1 = k0 + 1;
        v2f a;
        a.x = (m == k0) ? 0.25f : 0.0f;   // columns k0,k1 of 0.25*I16
        a.y = (m == k1) ? 0.25f : 0.0f;
        v2f b;
        b.x = p[(size_t)k0 * 16 + m];     // row k0 of tile, coalesced
        b.y = p[(size_t)k1 * 16 + m];
        c = __builtin_amdgcn_wmma_f32_16x16x4_f32(
                /*neg_a=*/false, a, /*neg_b=*/false, b,
                /*c_mod=*/(short)0, c, /*reuse_a=*/false, /*reuse_b=*/false);
    }
#pragma unroll
    for (int v = 0; v < 8; ++v) {
        const int row = v + 8 * hi;       // C/D layout: vgpr v, lane-half -> M
        p[(size_t)row * 16 + m] = c[v];
    }
}

// ---------------------------------------------------------------------
extern "C" void kernel_launch(void* const* d_in, const int* in_sizes, int n_in,
                              void* d_out, int out_size, void* d_ws, size_t ws_size,
                              hipStream_t stream) {
    const float* user_emb  = (const float*)d_in[0];
    const float* item_emb  = (const float*)d_in[1];
    const int*   edge_rows = (const int*)d_in[2];   // node ids < 2^31
    const int*   edge_cols = (const int*)d_in[3];
    const float* edge_vals = (const float*)d_in[4];
    float* out = (float*)d_out;                     // acc lives here

    const size_t nd  = (size_t)NUM_NODES * EMBED_DIM;   // 19.2M floats
    const size_t n4  = nd / 4;                          // 4.8M float4
    float* hA = (float*)d_ws;                           // ping
    float* hB = hA + nd;                                // pong (ws >= 153.6 MB)

    const int BLK = 256;
    const int gridN4 = (int)((n4 + BLK - 1) / BLK);                 // 18750

    lgcn_init_kernel<<<gridN4, BLK, 0, stream>>>(user_emb, item_emb, hA, out);
    lgcn_zero_kernel<<<gridN4, BLK, 0, stream>>>(hB, n4);

    float* src = hA;
    float* dst = hB;
    const long long waves = NUM_EDGES / EDGES_PER_WAVE;             // 9375
    const int spmmBlocks = (int)((waves * 32 + BLK - 1) / BLK);     // 1172

    for (int l = 0; l < NUM_LAYERS; ++l) {
        lgcn_spmm_kernel<<<spmmBlocks, BLK, 0, stream>>>(edge_rows, edge_cols,
                                                         edge_vals, src, dst);
        // acc += dst; zero the just-consumed src (it becomes next dst)
        float* z = (l < NUM_LAYERS - 1) ? src : (float*)nullptr;
        lgcn_add_zero_kernel<<<gridN4, BLK, 0, stream>>>(out, dst, z, n4);
        float* t = src; src = dst; dst = t;
    }

    // 75,000 tiles, 8 waves/block -> 9375 blocks exactly (EXEC all-1s per wave)
    const int tiles = (int)(nd / 256);
    lgcn_finalize_wmma_kernel<<<tiles / 8, BLK, 0, stream>>>(out);
}